// GCNEncoder_46694884442280
// MI455X (gfx1250) — compile-verified
//
#include <hip/hip_runtime.h>

typedef __attribute__((ext_vector_type(2))) float v2f;
typedef __attribute__((ext_vector_type(8))) float v8f;

#define N_NODES 100000
#define N_EDGES 3200000
#define IN_DIM  256
#define HID_DIM 64
#define OUT_DIM 64

// ---------------------------------------------------------------------------
// Degree / normalization kernels
// ---------------------------------------------------------------------------
__global__ void init_deg_kernel(float* __restrict__ deg, int n) {
    int i = blockIdx.x * blockDim.x + threadIdx.x;
    if (i < n) deg[i] = 1.0f;               // self-loop contributes 1
}

__global__ void deg_accum_kernel(const int* __restrict__ dst,
                                 float* __restrict__ deg, int ne) {
    int e = blockIdx.x * blockDim.x + threadIdx.x;
    if (e < ne) unsafeAtomicAdd(&deg[dst[e]], 1.0f);   // hw global_atomic_add_f32
}

__global__ void dis_kernel(float* __restrict__ deg, int n) {
    int i = blockIdx.x * blockDim.x + threadIdx.x;
    if (i < n) {
        float d = deg[i];
        deg[i] = (d > 0.0f) ? rsqrtf(d) : 0.0f;  // deg >= 1 always, kept for parity
    }
}

// ---------------------------------------------------------------------------
// fp32 WMMA GEMM:  C[M x N] = A[M x K] @ W[K x N]
// One block = 16 output rows, 4 waves = 4 col-tiles of 16 (requires N == 64).
// Requires M % 16 == 0 (100000 = 6250 * 16).
// Uses V_WMMA_F32_16X16X4_F32; A/W chunks staged in LDS with padded strides.
// ---------------------------------------------------------------------------
template <int K, int N>
__global__ __launch_bounds__(32 * (N / 16))
void gemm_wmma_f32(const float* __restrict__ A,
                   const float* __restrict__ W,
                   float* __restrict__ C) {
    constexpr int KB  = 64;          // K-chunk staged in LDS
    constexpr int LDA = KB + 4;      // pad: 16 rows hit distinct bank groups
    constexpr int LDW = N + 16;      // pad: half-wave K offsets land 32 banks apart
    constexpr int NT  = 32 * (N / 16);

    __shared__ float ldsA[16 * LDA];
    __shared__ float ldsW[KB * LDW];

    const int tid      = threadIdx.x;
    const int row_base = blockIdx.x * 16;
    const int warp     = tid >> 5;
    const int lane     = tid & 31;
    const int m        = lane & 15;   // row (A/C) or col (B/C) index within tile
    const int half     = lane >> 4;   // selects K pair / C row-half per ISA layout
    const int col      = warp * 16 + m;

    v8f acc = {0.f, 0.f, 0.f, 0.f, 0.f, 0.f, 0.f, 0.f};

    for (int kb = 0; kb < K; kb += KB) {
        __syncthreads();   // protect previous chunk's LDS reads
        // Stage A chunk: 16 x KB
        for (int idx = tid; idx < 16 * KB; idx += NT) {
            int r = idx >> 6;          // KB == 64
            int c = idx & 63;
            ldsA[r * LDA + c] = A[(row_base + r) * K + kb + c];
        }
        // Stage W chunk: KB x N
        for (int idx = tid; idx < KB * N; idx += NT) {
            int r = idx / N;
            int c = idx % N;
            ldsW[r * LDW + c] = W[(kb + r) * N + c];
        }
        __syncthreads();

        const float* arow = &ldsA[m * LDA];
#pragma unroll
        for (int k = 0; k < KB; k += 4) {
            // A 16x4 layout: VGPR0/1 = K {half*2, half*2+1}, row = lane%16
            v2f a, b;
            a.x = arow[k + half * 2];
            a.y = arow[k + half * 2 + 1];
            // B 4x16 layout mirrors A: K = half*2 + j, col = lane%16
            b.x = ldsW[(k + half * 2) * LDW + col];
            b.y = ldsW[(k + half * 2 + 1) * LDW + col];
            acc = __builtin_amdgcn_wmma_f32_16x16x4_f32(
                false, a, false, b, (short)0, acc, false, false);
        }
    }

    // C 16x16 f32 layout: VGPR v -> row = v + 8*half, col = lane%16
#pragma unroll
    for (int v = 0; v < 8; ++v)
        C[(row_base + v + half * 8) * N + col] = acc[v];
}

// ---------------------------------------------------------------------------
// Aggregation kernels (feature dim fixed at 64)
// ---------------------------------------------------------------------------
// out[i,:] = dis[i]^2 * h[i,:]   (self-loop term; also initializes out)
__global__ void selfloop_init_kernel(const float* __restrict__ h,
                                     const float* __restrict__ dis,
                                     float* __restrict__ out, int n) {
    int t = blockIdx.x * blockDim.x + threadIdx.x;
    if (t < n * 64) {
        float s = dis[t >> 6];
        out[t] = s * s * h[t];
    }
}

// One wave per edge; lane f handles features f and f+32.
__global__ void edge_scatter_kernel(const float* __restrict__ h,
                                    const int* __restrict__ src,
                                    const int* __restrict__ dst,
                                    const float* __restrict__ dis,
                                    float* __restrict__ out, int ne) {
    int e    = blockIdx.x * (blockDim.x >> 5) + (threadIdx.x >> 5);
    int lane = threadIdx.x & 31;
    if (e >= ne) return;
    int   s    = src[e];
    int   d    = dst[e];
    float norm = dis[s] * dis[d];
    float v0 = norm * h[s * 64 + lane];
    float v1 = norm * h[s * 64 + 32 + lane];
    unsafeAtomicAdd(&out[d * 64 + lane], v0);
    unsafeAtomicAdd(&out[d * 64 + 32 + lane], v1);
}

__global__ void bias_act_kernel(float* __restrict__ y,
                                const float* __restrict__ b,
                                int n, int relu) {
    int t = blockIdx.x * blockDim.x + threadIdx.x;
    if (t < n * 64) {
        float v = y[t] + b[t & 63];
        if (relu) v = fmaxf(v, 0.0f);
        y[t] = v;
    }
}

// ---------------------------------------------------------------------------
// Launch
// ---------------------------------------------------------------------------
extern "C" void kernel_launch(void* const* d_in, const int* in_sizes, int n_in,
                              void* d_out, int out_size, void* d_ws, size_t ws_size,
                              hipStream_t stream) {
    const float* x  = (const float*)d_in[0];
    const int*   ei = (const int*)  d_in[1];
    const float* W1 = (const float*)d_in[2];
    const float* b1 = (const float*)d_in[3];
    const float* W2 = (const float*)d_in[4];
    const float* b2 = (const float*)d_in[5];
    float* out = (float*)d_out;

    const int* src = ei;             // edge_index[0]
    const int* dst = ei + N_EDGES;   // edge_index[1]

    // Workspace layout (floats): dis | h1 | agg1 ; h2 reuses h1's slot.
    float* ws   = (float*)d_ws;
    float* dis  = ws;                           // 100000 (padded region 102400)
    float* h1   = ws + 102400;                  // 100000 * 64
    float* agg1 = h1 + (size_t)N_NODES * 64;    // 100000 * 64
    float* h2   = h1;                           // h1 dead after layer-1 scatter

    const int n  = N_NODES;
    const int nf = n * 64;

    // --- normalization: deg -> deg_inv_sqrt (in place in `dis`)
    init_deg_kernel<<<(n + 255) / 256, 256, 0, stream>>>(dis, n);
    deg_accum_kernel<<<(N_EDGES + 255) / 256, 256, 0, stream>>>(dst, dis, N_EDGES);
    dis_kernel<<<(n + 255) / 256, 256, 0, stream>>>(dis, n);

    // --- layer 1: h1 = x @ W1 ; agg1 = scatter(norm * h1[src]) ; relu(+b1)
    gemm_wmma_f32<IN_DIM, HID_DIM><<<n / 16, 128, 0, stream>>>(x, W1, h1);
    selfloop_init_kernel<<<(nf + 255) / 256, 256, 0, stream>>>(h1, dis, agg1, n);
    edge_scatter_kernel<<<(N_EDGES + 7) / 8, 256, 0, stream>>>(h1, src, dst, dis,
                                                               agg1, N_EDGES);
    bias_act_kernel<<<(nf + 255) / 256, 256, 0, stream>>>(agg1, b1, n, 1);

    // --- layer 2: h2 = agg1 @ W2 ; out = scatter(norm * h2[src]) + b2
    gemm_wmma_f32<HID_DIM, OUT_DIM><<<n / 16, 128, 0, stream>>>(agg1, W2, h2);
    selfloop_init_kernel<<<(nf + 255) / 256, 256, 0, stream>>>(h2, dis, out, n);
    edge_scatter_kernel<<<(N_EDGES + 7) / 8, 256, 0, stream>>>(h2, src, dst, dis,
                                                               out, N_EDGES);
    bias_act_kernel<<<(nf + 255) / 256, 256, 0, stream>>>(out, b2, n, 0);
}